// ShortConv_2937757630927
// MI455X (gfx1250) — compile-verified
//
#include <hip/hip_runtime.h>
#include <cstdint>
#include <cstddef>

typedef float v4f __attribute__((ext_vector_type(4)));

// Problem constants (from reference setup_inputs)
constexpr int Bb = 2;
constexpr int Ll = 4096;
constexpr int Gg = 4;
constexpr int Dd = 2048;
constexpr int Kk = 4;

constexpr int THREADS = 256;          // 8 wave32s
constexpr int PT      = Dd / THREADS; // 8 floats per thread per row
constexpr int T_TILE  = 64;           // timesteps per block
constexpr int NBUF    = 4;            // LDS row ring buffers (32 KB)
constexpr int LA      = 2;            // async lookahead (rows)
constexpr float EPSf  = 1e-5f;

__global__ __launch_bounds__(THREADS) void shortconv_fused_kernel(
    const float* __restrict__ x,       // [B, L, G, D]
    const float* __restrict__ conv_w,  // [G*D, 1, K]
    const float* __restrict__ norm_w,  // [G, D]
    float* __restrict__ out)           // [B, L, G, D]
{
    __shared__ float smem[NBUF][Dd];       // row staging (raw x rows)
    __shared__ float red[THREADS / 32];    // per-wave partial sums

    const int tid  = threadIdx.x;
    const int wave = tid >> 5;
    const int lane = tid & 31;

    constexpr int NCH = Ll / T_TILE;
    const int chunk = blockIdx.x % NCH;
    const int bg    = blockIdx.x / NCH;
    const int g     = bg % Gg;
    const int b     = bg / Gg;
    const int t0    = chunk * T_TILE;

    const int d0 = tid * PT;          // this thread's channel slice within group
    const int c0 = g * Dd + d0;       // global channel index base

    // ---- preload per-thread weights (all 16B-aligned) ----
    float nw[PT];
    #pragma unroll
    for (int j = 0; j < PT; j += 4) {
        v4f q = *(const v4f*)&norm_w[c0 + j];
        nw[j + 0] = q.x; nw[j + 1] = q.y; nw[j + 2] = q.z; nw[j + 3] = q.w;
    }
    float cw[PT][Kk];
    #pragma unroll
    for (int j = 0; j < PT; ++j) {
        v4f q = *(const v4f*)&conv_w[(size_t)(c0 + j) * Kk];
        cw[j][0] = q.x; cw[j][1] = q.y; cw[j][2] = q.z; cw[j][3] = q.w;
    }

    // register ring of normalized rows: r0 = newest (t), r3 = oldest (t-3)
    float r0[PT], r1[PT], r2[PT], r3[PT];
    #pragma unroll
    for (int j = 0; j < PT; ++j) { r0[j] = r1[j] = r2[j] = r3[j] = 0.0f; }

    // Each wave async-copies its 1KB slice of a row: 2 x b128 (512B each).
    const uint32_t voff = (uint32_t)(wave * 1024 + lane * 16);

    auto issue_row = [&](int r) {
        if (r < 0 || r >= Ll) return;  // uniform branch
        const float* rowp = x + (((size_t)b * Ll + r) * Gg + g) * (size_t)Dd;
        const uint64_t sa  = (uint64_t)(uintptr_t)rowp;
        const uint32_t lds = (uint32_t)(uintptr_t)(&smem[r & (NBUF - 1)][0]) + voff;
        // GVS mode: mem = SGPR base + VGPR offset + ioffset; LDS = vdst + ioffset
        asm volatile("global_load_async_to_lds_b128 %0, %1, %2 offset:0"
                     :: "v"(lds), "v"(voff), "s"(sa) : "memory");
        asm volatile("global_load_async_to_lds_b128 %0, %1, %2 offset:512"
                     :: "v"(lds), "v"(voff), "s"(sa) : "memory");
    };

    // prologue: prefetch rows for steps 0 .. LA-1
    issue_row(t0 - (Kk - 1) + 0);
    issue_row(t0 - (Kk - 1) + 1);

    constexpr int S = T_TILE + Kk - 1;     // halo steps + output steps
    for (int s = 0; s < S; ++s) {
        const int r = t0 - (Kk - 1) + s;   // row processed this step

        // row r must be resident; allow LA-1 rows (2 instrs/wave) in flight
        asm volatile("s_wait_asynccnt 2" ::: "memory");
        __syncthreads();   // row r visible to all waves; red[] free for reuse

        // ---- load slice from LDS, accumulate sum of squares ----
        float v[PT];
        float ss = 0.0f;
        if (r >= 0) {
            const float* rowlds = &smem[r & (NBUF - 1)][d0];
            #pragma unroll
            for (int j = 0; j < PT; j += 4) {
                v4f q = *(const v4f*)&rowlds[j];
                v[j + 0] = q.x; v[j + 1] = q.y; v[j + 2] = q.z; v[j + 3] = q.w;
                ss += q.x * q.x + q.y * q.y + q.z * q.z + q.w * q.w;
            }
        } else {
            #pragma unroll
            for (int j = 0; j < PT; ++j) v[j] = 0.0f;
        }

        // ---- block-wide reduction (wave32 shuffle + LDS scratch) ----
        #pragma unroll
        for (int m = 16; m >= 1; m >>= 1) ss += __shfl_xor(ss, m, 32);
        if (lane == 0) red[wave] = ss;
        __syncthreads();
        float tot = 0.0f;
        #pragma unroll
        for (int wv = 0; wv < THREADS / 32; ++wv) tot += red[wv];

        const float scale = (r >= 0) ? rsqrtf(tot * (1.0f / Dd) + EPSf) : 0.0f;

        // ---- shift register ring, insert normalized row ----
        #pragma unroll
        for (int j = 0; j < PT; ++j) {
            r3[j] = r2[j]; r2[j] = r1[j]; r1[j] = r0[j];
            r0[j] = v[j] * scale * nw[j];
        }

        // ---- depthwise conv + SiLU + store (after halo is primed) ----
        if (s >= Kk - 1) {
            float* op = out + (((size_t)b * Ll + r) * Gg + g) * (size_t)Dd + d0;
            #pragma unroll
            for (int j = 0; j < PT; j += 4) {
                v4f o;
                #pragma unroll
                for (int u = 0; u < 4; ++u) {
                    const int jj = j + u;
                    float a = cw[jj][0] * r3[jj] + cw[jj][1] * r2[jj]
                            + cw[jj][2] * r1[jj] + cw[jj][3] * r0[jj];
                    float sg = 1.0f / (1.0f + __expf(-a));
                    o[u] = a * sg;
                }
                __builtin_nontemporal_store(o, (v4f*)(op + j));
            }
        }

        // ---- prefetch row r+LA into its ring slot ----
        if (s + LA < S) issue_row(r + LA);
    }
}

extern "C" void kernel_launch(void* const* d_in, const int* in_sizes, int n_in,
                              void* d_out, int out_size, void* d_ws, size_t ws_size,
                              hipStream_t stream) {
    (void)in_sizes; (void)n_in; (void)out_size; (void)d_ws; (void)ws_size;
    const float* x      = (const float*)d_in[0];
    const float* conv_w = (const float*)d_in[1];
    const float* norm_w = (const float*)d_in[2];
    float* out          = (float*)d_out;

    dim3 grid(Bb * Gg * (Ll / T_TILE));   // 512 blocks
    dim3 block(THREADS);                  // 8 wave32s
    shortconv_fused_kernel<<<grid, block, 0, stream>>>(x, conv_w, norm_w, out);
}